// GCN_32633161515373
// MI455X (gfx1250) — compile-verified
//
#include <hip/hip_runtime.h>

typedef __attribute__((ext_vector_type(2))) float v2f;
typedef __attribute__((ext_vector_type(8))) float v8f;

#define HCH 128   // hidden channels
#define NG  512   // number of graphs
#define EPSV 1e-5f

// ---------------------------------------------------------------- CDNA5 async copy
// One b32 moved from global memory straight into LDS (no VGPR data), tracked
// by ASYNCcnt. lds_byte_addr is the low-32 bits of a generic pointer to a
// __shared__ object (== LDS byte offset per the flat-aperture rules).
__device__ __forceinline__ void async_g2lds_b32(const float* g, unsigned lds_byte_addr) {
  asm volatile("global_load_async_to_lds_b32 %0, %1, off"
               :: "v"(lds_byte_addr), "v"(g)
               : "memory");
}
__device__ __forceinline__ void wait_asynccnt0() {
  asm volatile("s_wait_asynccnt 0" ::: "memory");
}

// ---------------------------------------------------------------- utilities
__global__ void fill_kernel(float* __restrict__ p, float v, int n) {
  int i = blockIdx.x * blockDim.x + threadIdx.x;
  if (i < n) p[i] = v;
}

__global__ void deg_edges_kernel(const int* __restrict__ dst, float* __restrict__ deg, int E) {
  int e = blockIdx.x * blockDim.x + threadIdx.x;
  if (e < E) atomicAdd(&deg[dst[e]], 1.0f);
}

__global__ void counts_kernel(const int* __restrict__ batch, float* __restrict__ counts, int N) {
  int i = blockIdx.x * blockDim.x + threadIdx.x;
  if (i < N) atomicAdd(&counts[batch[i]], 1.0f);
}

__global__ void rsqrt_inplace_kernel(float* __restrict__ p, int n) {
  int i = blockIdx.x * blockDim.x + threadIdx.x;
  if (i < n) p[i] = rsqrtf(p[i]);   // deg >= 1 always (self-loops)
}

// ---------------------------------------------------------------- WMMA GEMM
// C[nrows, 128] = A[nrows, K] @ W[K, 128]  (+ bias if non-null), fp32 via
// V_WMMA_F32_16X16X4_F32. Block = 8 waves; wave w computes rows
// [blk*128 + w*16, +16) over all 128 columns (8 16x16 tiles).
// W staged into LDS with async global->LDS copies in a k-pair-interleaved
// layout so each lane's B fragment is one aligned ds_load_b64:
//   element (k, col) lives at sW[((k>>1)*HCH + col)*2 + (k&1)].
template <int K>
__launch_bounds__(256)
__global__ void gemm_wmma_kernel(const float* __restrict__ A,
                                 const float* __restrict__ W,
                                 const float* __restrict__ bias,
                                 float* __restrict__ C, int nrows) {
  __shared__ float sW[K * HCH];

  for (int i = threadIdx.x; i < K * HCH; i += 256) {
    const int k = i >> 7;          // row of W (source is row-major [K,128])
    const int col = i & (HCH - 1);
    const int didx = (((k >> 1) * HCH + col) << 1) | (k & 1);
    async_g2lds_b32(W + i, (unsigned)(size_t)&sW[didx]);
  }
  wait_asynccnt0();
  __syncthreads();

  const int wave = threadIdx.x >> 5;
  const int lane = threadIdx.x & 31;
  const int half = lane >> 4;      // 0: K=kb,kb+1 / M=r ; 1: K=kb+2,kb+3 / M=r+8
  const int lr   = lane & 15;
  const int m0   = blockIdx.x * 128 + wave * 16;
  if (m0 >= nrows) return;

  const v8f vzero = {0.f, 0.f, 0.f, 0.f, 0.f, 0.f, 0.f, 0.f};
  v8f acc[8];
#pragma unroll
  for (int t = 0; t < 8; ++t) acc[t] = vzero;

  int arow = m0 + lr;
  if (arow >= nrows) arow = nrows - 1;   // clamp; garbage rows never stored
  const float* __restrict__ Ap = A + (size_t)arow * K;

  for (int k0 = 0; k0 < K; k0 += 4) {
    const int kb = k0 + 2 * half;                 // even
    const v2f a = *(const v2f*)&Ap[kb];           // global_load_b64
    const float* __restrict__ sp = &sW[(kb >> 1) * (HCH * 2)];
#pragma unroll
    for (int t = 0; t < 8; ++t) {
      const int col = t * 16 + lr;
      const v2f b = *(const v2f*)&sp[col * 2];    // single ds_load_b64
      acc[t] = __builtin_amdgcn_wmma_f32_16x16x4_f32(
          /*neg_a=*/false, a, /*neg_b=*/false, b,
          /*c_mod=*/(short)0, acc[t], /*reuse_a=*/false, /*reuse_b=*/false);
    }
  }

#pragma unroll
  for (int t = 0; t < 8; ++t) {
    const int col = t * 16 + lr;
    const float badd = bias ? bias[col] : 0.0f;
#pragma unroll
    for (int r = 0; r < 8; ++r) {
      const int row = m0 + r + 8 * half;
      if (row < nrows) C[(size_t)row * HCH + col] = acc[t][r] + badd;
    }
  }
}

// ---------------------------------------------------------------- GCN aggregation
// agg[i][c] = tmp[i][c] * dis[i]^2 + bias[c]   (self-loop term + bias as init)
__global__ void agg_init_kernel(const float* __restrict__ tmp,
                                const float* __restrict__ dis,
                                const float* __restrict__ bias,
                                float* __restrict__ agg, int N) {
  int idx = blockIdx.x * blockDim.x + threadIdx.x;
  if (idx >= N * HCH) return;
  int i = idx >> 7, c = idx & (HCH - 1);
  float d = dis[i];
  agg[idx] = tmp[idx] * d * d + bias[c];
}

// one wave per edge; lanes cover 128 channels (4 each, stride 32 -> coalesced)
__global__ void agg_edges_kernel(const float* __restrict__ tmp,
                                 const int* __restrict__ src,
                                 const int* __restrict__ dst,
                                 const float* __restrict__ dis,
                                 float* __restrict__ agg, int E) {
  int e = blockIdx.x * 8 + (threadIdx.x >> 5);
  if (e >= E) return;
  int lane = threadIdx.x & 31;
  int s = src[e], d = dst[e];
  float nrm = dis[s] * dis[d];
  const float* __restrict__ trow = tmp + (size_t)s * HCH;
  float* __restrict__ arow = agg + (size_t)d * HCH;
#pragma unroll
  for (int j = 0; j < 4; ++j) {
    int ch = lane + 32 * j;
    atomicAdd(arow + ch, trow[ch] * nrm);
  }
}

// ---------------------------------------------------------------- graph layernorm
__global__ void gstats_kernel(const float* __restrict__ h, const int* __restrict__ batch,
                              float* __restrict__ gsum, float* __restrict__ gsumsq, int N) {
  int i = blockIdx.x * 8 + (threadIdx.x >> 5);
  if (i >= N) return;
  int lane = threadIdx.x & 31;
  const float* __restrict__ row = h + (size_t)i * HCH;
  float s = 0.f, ss = 0.f;
#pragma unroll
  for (int j = 0; j < 4; ++j) {
    float v = row[lane + 32 * j];
    s += v;
    ss += v * v;
  }
#pragma unroll
  for (int off = 16; off > 0; off >>= 1) {
    s  += __shfl_xor(s,  off, 32);
    ss += __shfl_xor(ss, off, 32);
  }
  if (lane == 0) {
    int g = batch[i];
    atomicAdd(&gsum[g], s);
    atomicAdd(&gsumsq[g], ss);
  }
}

__global__ void gfinalize_kernel(const float* __restrict__ gsum,
                                 const float* __restrict__ gsumsq,
                                 const float* __restrict__ counts,
                                 float* __restrict__ mean, float* __restrict__ inv, int G) {
  int g = blockIdx.x * blockDim.x + threadIdx.x;
  if (g >= G) return;
  float denom = fmaxf(counts[g] * (float)HCH, 1.0f);
  float m = gsum[g] / denom;
  float var = fmaxf(gsumsq[g] / denom - m * m, 0.0f);
  mean[g] = m;
  inv[g] = rsqrtf(var + EPSV);
}

__global__ void norm_relu_kernel(float* __restrict__ h, const int* __restrict__ batch,
                                 const float* __restrict__ mean, const float* __restrict__ inv,
                                 const float* __restrict__ gamma, const float* __restrict__ beta,
                                 int N) {
  int idx = blockIdx.x * blockDim.x + threadIdx.x;
  if (idx >= N * HCH) return;
  int i = idx >> 7, c = idx & (HCH - 1);
  int g = batch[i];
  float v = (h[idx] - mean[g]) * inv[g] * gamma[c] + beta[c];
  h[idx] = fmaxf(v, 0.0f);
}

// ---------------------------------------------------------------- pooling + fc2
__global__ void pool_accum_kernel(const float* __restrict__ h, const int* __restrict__ batch,
                                  float* __restrict__ pooled, int N) {
  int idx = blockIdx.x * blockDim.x + threadIdx.x;
  if (idx >= N * HCH) return;
  int i = idx >> 7, c = idx & (HCH - 1);
  atomicAdd(&pooled[(size_t)batch[i] * HCH + c], h[idx]);
}

__global__ void pool_out_kernel(const float* __restrict__ pooled,
                                const float* __restrict__ counts,
                                const float* __restrict__ w, const float* __restrict__ b,
                                float* __restrict__ out, int G) {
  int g = blockIdx.x * 8 + (threadIdx.x >> 5);
  if (g >= G) return;
  int lane = threadIdx.x & 31;
  const float* __restrict__ row = pooled + (size_t)g * HCH;
  float acc = 0.f;
#pragma unroll
  for (int j = 0; j < 4; ++j) {
    int ch = lane + 32 * j;
    acc += row[ch] * w[ch];
  }
#pragma unroll
  for (int off = 16; off > 0; off >>= 1) acc += __shfl_xor(acc, off, 32);
  if (lane == 0) out[g] = acc / fmaxf(counts[g], 1.0f) + b[0];
}

// ---------------------------------------------------------------- driver
extern "C" void kernel_launch(void* const* d_in, const int* in_sizes, int n_in,
                              void* d_out, int out_size, void* d_ws, size_t ws_size,
                              hipStream_t stream) {
  const float* x      = (const float*)d_in[0];
  const int*   ei     = (const int*)  d_in[1];
  const int*   batch  = (const int*)  d_in[2];
  const float* fc1_w  = (const float*)d_in[3];
  const float* fc1_b  = (const float*)d_in[4];
  const float* gc1_w  = (const float*)d_in[5];
  const float* gc1_b  = (const float*)d_in[6];
  const float* gn1_g  = (const float*)d_in[7];
  const float* gn1_b  = (const float*)d_in[8];
  const float* gc2_w  = (const float*)d_in[9];
  const float* gc2_b  = (const float*)d_in[10];
  const float* gn2_g  = (const float*)d_in[11];
  const float* gn2_b  = (const float*)d_in[12];
  const float* fc2_w  = (const float*)d_in[13];
  const float* fc2_b  = (const float*)d_in[14];
  float* out = (float*)d_out;

  const int N = in_sizes[0] / 64;   // F_IN = 64
  const int E = in_sizes[1] / 2;
  const int G = NG;

  const int* src = ei;          // edge_index[0]
  const int* dst = ei + E;      // edge_index[1]

  // workspace carve-up (all fp32)
  float* p = (float*)d_ws;
  float* deg    = p;  p += N;            // becomes deg_inv_sqrt in place
  float* counts = p;  p += G;
  float* gsum   = p;  p += G;
  float* gsumsq = p;  p += G;
  float* gmean  = p;  p += G;
  float* ginv   = p;  p += G;
  float* pooled = p;  p += (size_t)G * HCH;
  float* bufA   = p;  p += (size_t)N * HCH;
  float* bufB   = p;  p += (size_t)N * HCH;
  float* bufC   = p;  /* p += N*HCH */

  const int TB = 256;
  const int gridNH   = (N * HCH + TB - 1) / TB;
  const int gridN    = (N + TB - 1) / TB;
  const int gridE    = (E + TB - 1) / TB;
  const int gridE8   = (E + 7) / 8;          // wave-per-edge
  const int gridN8   = (N + 7) / 8;          // wave-per-node
  const int gridG    = (G + TB - 1) / TB;
  const int gridG8   = (G + 7) / 8;
  const int gridGemm = (N + 127) / 128;

  // degrees (self-loops folded in as init=1) and per-graph node counts
  fill_kernel<<<gridN, TB, 0, stream>>>(deg, 1.0f, N);
  fill_kernel<<<gridG, TB, 0, stream>>>(counts, 0.0f, G);
  deg_edges_kernel<<<gridE, TB, 0, stream>>>(dst, deg, E);
  counts_kernel<<<gridN, TB, 0, stream>>>(batch, counts, N);
  rsqrt_inplace_kernel<<<gridN, TB, 0, stream>>>(deg, N);
  const float* dis = deg;

  // fc1: bufA = x @ fc1_w + fc1_b
  gemm_wmma_kernel<64><<<gridGemm, TB, 0, stream>>>(x, fc1_w, fc1_b, bufA, N);

  // --- layer 1: gcn_conv -> graph LN -> relu  (h1 ends up in bufC)
  gemm_wmma_kernel<128><<<gridGemm, TB, 0, stream>>>(bufA, gc1_w, nullptr, bufB, N);
  agg_init_kernel<<<gridNH, TB, 0, stream>>>(bufB, dis, gc1_b, bufC, N);
  agg_edges_kernel<<<gridE8, TB, 0, stream>>>(bufB, src, dst, dis, bufC, E);
  fill_kernel<<<gridG, TB, 0, stream>>>(gsum, 0.0f, G);
  fill_kernel<<<gridG, TB, 0, stream>>>(gsumsq, 0.0f, G);
  gstats_kernel<<<gridN8, TB, 0, stream>>>(bufC, batch, gsum, gsumsq, N);
  gfinalize_kernel<<<gridG, TB, 0, stream>>>(gsum, gsumsq, counts, gmean, ginv, G);
  norm_relu_kernel<<<gridNH, TB, 0, stream>>>(bufC, batch, gmean, ginv, gn1_g, gn1_b, N);

  // --- layer 2 (h2 ends up in bufB)
  gemm_wmma_kernel<128><<<gridGemm, TB, 0, stream>>>(bufC, gc2_w, nullptr, bufA, N);
  agg_init_kernel<<<gridNH, TB, 0, stream>>>(bufA, dis, gc2_b, bufB, N);
  agg_edges_kernel<<<gridE8, TB, 0, stream>>>(bufA, src, dst, dis, bufB, E);
  fill_kernel<<<gridG, TB, 0, stream>>>(gsum, 0.0f, G);
  fill_kernel<<<gridG, TB, 0, stream>>>(gsumsq, 0.0f, G);
  gstats_kernel<<<gridN8, TB, 0, stream>>>(bufB, batch, gsum, gsumsq, N);
  gfinalize_kernel<<<gridG, TB, 0, stream>>>(gsum, gsumsq, counts, gmean, ginv, G);
  norm_relu_kernel<<<gridNH, TB, 0, stream>>>(bufB, batch, gmean, ginv, gn2_g, gn2_b, N);

  // --- mean pool + fc2
  fill_kernel<<<(G * HCH + TB - 1) / TB, TB, 0, stream>>>(pooled, 0.0f, G * HCH);
  pool_accum_kernel<<<gridNH, TB, 0, stream>>>(bufB, batch, pooled, N);
  pool_out_kernel<<<gridG8, TB, 0, stream>>>(pooled, counts, fc2_w, fc2_b, out, G);

  (void)n_in; (void)out_size; (void)ws_size;
}